// LocalizedAttention_59871844106332
// MI455X (gfx1250) — compile-verified
//
#include <hip/hip_runtime.h>

typedef __attribute__((ext_vector_type(2))) float v2f;
typedef __attribute__((ext_vector_type(8))) float v8f;

#define BATCH 2
#define CIN   128
#define VOL   32768          // 32*32*32
#define NCH   160            // q(16) + k(16) + v(128), channel-last in ws

// ---------------------------------------------------------------------------
// Kernel 1: fused Q/K/V 1x1x1 conv as an fp32 WMMA GEMM.
//   ws[vox][m] = sum_k Wcat[m,k] * x[k,vox] + bcat[m]
// Wcat rows: [0,16)=wq, [16,32)=wk, [32,160)=wv.
// One wave = one full 32-voxel x-row (two 16-column tiles sharing the same
// A fragments), all 10 M-tiles of 16 channels, K=128 swept 4 at a time with
// V_WMMA_F32_16X16X4_F32 (f32 in, f32 accumulate -> matches fp32 reference).
// Per k-step: 10 A loads (b64) + 4 B loads (b32) feed 20 WMMAs.
// ---------------------------------------------------------------------------
__global__ __launch_bounds__(128) void qkv_gemm_kernel(
    const float* __restrict__ x,
    const float* __restrict__ wq, const float* __restrict__ bq,
    const float* __restrict__ wk, const float* __restrict__ bk,
    const float* __restrict__ wv, const float* __restrict__ bv,
    float* __restrict__ ws)
{
  const int lane = threadIdx.x & 31;
  const int wave = blockIdx.x * (blockDim.x >> 5) + (threadIdx.x >> 5);
  // row decode: 32 y, 32 z, 2 batches -> 2048 waves total
  const int y = wave & 31;
  const int z = (wave >> 5) & 31;
  const int b = wave >> 10;

  const int ln   = lane & 15;   // N index (voxel within tile) / M row for A
  const int half = lane >> 4;   // selects K pair {0,1} vs {2,3}

  const float* xb = x + (size_t)b * CIN * VOL + z * 1024 + y * 32;

  v8f acc0[10] = {};            // columns x = 0..15
  v8f acc1[10] = {};            // columns x = 16..31

  for (int k0 = 0; k0 < CIN; k0 += 4) {
    const int ksel = k0 + half * 2;
    // B fragments: 4x16 tiles of x. VGPR0 = rows {0,2}, VGPR1 = rows {1,3}.
    const float* xr0 = xb + (size_t)ksel * VOL;
    const float* xr1 = xb + (size_t)(ksel + 1) * VOL;
    v2f bm0, bm1;
    bm0.x = xr0[ln];       bm0.y = xr1[ln];
    bm1.x = xr0[ln + 16];  bm1.y = xr1[ln + 16];
#pragma unroll
    for (int t = 0; t < 10; ++t) {
      // A fragment: 16x4 tile of weights; lane row = ln, two consecutive k's.
      const float* wrow =
          (t == 0) ? (wq + ln * CIN) :
          (t == 1) ? (wk + ln * CIN) :
                     (wv + (size_t)((t - 2) * 16 + ln) * CIN);
      v2f am = *(const v2f*)(wrow + ksel);
      acc0[t] = __builtin_amdgcn_wmma_f32_16x16x4_f32(
          false, am, false, bm0, (short)0, acc0[t], false, false);
      acc1[t] = __builtin_amdgcn_wmma_f32_16x16x4_f32(
          false, am, false, bm1, (short)0, acc1[t], false, false);
    }
  }

  // Store channel-last: ws[((b,z,y,x))*160 + m], add bias.
  const size_t vi0 = (size_t)(((b * 32 + z) * 32 + y)) * 32;
  float* outp0 = ws + (vi0 + ln) * NCH;        // C/D layout: lane -> column N=ln
  float* outp1 = ws + (vi0 + 16 + ln) * NCH;
#pragma unroll
  for (int t = 0; t < 10; ++t) {
    const float* bptr = (t == 0) ? bq : (t == 1) ? bk : (bv + (t - 2) * 16);
#pragma unroll
    for (int r = 0; r < 8; ++r) {
      const int ml = r + 8 * half;       // row within 16-row tile for this lane
      const float bias = bptr[ml];
      outp0[t * 16 + ml] = acc0[t][r] + bias;
      outp1[t * 16 + ml] = acc1[t][r] + bias;
    }
  }
}

// ---------------------------------------------------------------------------
// Kernel 2: windowed attention. One 128-thread block per voxel.
// Out-of-range neighbors use k=bk, v=bv (== conv of zero padding + bias),
// faithful to the reference's padded-conv semantics.
// ---------------------------------------------------------------------------
__global__ __launch_bounds__(128) void attn_kernel(
    const float* __restrict__ x,
    const float* __restrict__ bk, const float* __restrict__ bv,
    const float* __restrict__ gamma,
    const float* __restrict__ ws,
    float* __restrict__ out)
{
  __shared__ float s_q[16];
  __shared__ float s_bk[16];
  __shared__ float s_e[27];
  __shared__ float s_inv;

  const int vid = blockIdx.x;            // == linear interior voxel index
  const int xx = vid & 31;
  const int yy = (vid >> 5) & 31;
  const int zz = (vid >> 10) & 31;
  const int b  = vid >> 15;
  const int tid = threadIdx.x;
  const size_t vi = (size_t)vid;

  if (tid < 16) { s_q[tid] = ws[vi * NCH + tid]; s_bk[tid] = bk[tid]; }
  __syncthreads();

  // energies: 27 threads, dot(q, k_neighbor) over 16 channels
  if (tid < 27) {
    const int dz = tid / 9 - 1, dy = (tid / 3) % 3 - 1, dx = tid % 3 - 1;
    const int nz = zz + dz, ny = yy + dy, nx = xx + dx;
    const bool inb = (unsigned)nz < 32u && (unsigned)ny < 32u && (unsigned)nx < 32u;
    float e = 0.f;
    if (inb) {
      const float* kp =
          ws + (size_t)(((b * 32 + nz) * 32 + ny) * 32 + nx) * NCH + 16;
#pragma unroll
      for (int c = 0; c < 16; ++c) e += s_q[c] * kp[c];
    } else {
#pragma unroll
      for (int c = 0; c < 16; ++c) e += s_q[c] * s_bk[c];
    }
    s_e[tid] = e;
  }
  __syncthreads();

  // softmax (27-wide, serial on one lane is cheap)
  if (tid == 0) {
    float m = s_e[0];
    for (int n = 1; n < 27; ++n) m = fmaxf(m, s_e[n]);
    float s = 0.f;
    for (int n = 0; n < 27; ++n) { float t = __expf(s_e[n] - m); s_e[n] = t; s += t; }
    s_inv = 1.f / s;
  }
  __syncthreads();

  // weighted sum of v: thread == channel
  const int c = tid;
  const float bvc = bv[c];
  float local = 0.f;
#pragma unroll
  for (int n = 0; n < 27; ++n) {
    const int dz = n / 9 - 1, dy = (n / 3) % 3 - 1, dx = n % 3 - 1;
    const int nz = zz + dz, ny = yy + dy, nx = xx + dx;
    const bool inb = (unsigned)nz < 32u && (unsigned)ny < 32u && (unsigned)nx < 32u;
    const float v = inb
        ? ws[(size_t)(((b * 32 + nz) * 32 + ny) * 32 + nx) * NCH + 32 + c]
        : bvc;
    local += s_e[n] * v;
  }

  const size_t oidx = ((size_t)b * CIN + c) * VOL + zz * 1024 + yy * 32 + xx;
  out[oidx] = gamma[0] * local * s_inv + x[oidx];
}

extern "C" void kernel_launch(void* const* d_in, const int* in_sizes, int n_in,
                              void* d_out, int out_size, void* d_ws, size_t ws_size,
                              hipStream_t stream) {
  const float* x     = (const float*)d_in[0];
  const float* wq    = (const float*)d_in[1];
  const float* bq    = (const float*)d_in[2];
  const float* wk    = (const float*)d_in[3];
  const float* bk    = (const float*)d_in[4];
  const float* wv    = (const float*)d_in[5];
  const float* bv    = (const float*)d_in[6];
  const float* gamma = (const float*)d_in[7];
  float* ws  = (float*)d_ws;   // needs NCH * B*VOL * 4 = ~42 MB (fits in L2)
  float* out = (float*)d_out;

  // 2048 row-waves, 4 waves per block -> 512 blocks
  qkv_gemm_kernel<<<512, 128, 0, stream>>>(x, wq, bq, wk, bk, wv, bv, ws);
  // one block per interior voxel
  attn_kernel<<<BATCH * VOL, 128, 0, stream>>>(x, bk, bv, gamma, ws, out);
}